// NounLoss_26216480375254
// MI455X (gfx1250) — compile-verified
//
#include <hip/hip_runtime.h>

typedef __attribute__((ext_vector_type(2))) float v2f;
typedef __attribute__((ext_vector_type(8))) float v8f;

#define FDIM 2
#define BDIM 64
#define SEQ  512
#define DIM  768
#define PAIRS (FDIM * BDIM)          // 128
#define TILES (SEQ / 16)             // 32 tiles of 16 tokens per (f,b)
#define WAVES_PER_WG 8               // 256 threads = 8 wave32
#define WGS_PER_PAIR 4               // 4 WGs * 8 waves = 32 tiles

// One wave computes dot products for a 16-token tile of one (f,b) pair via
// V_WMMA_F32_16X16X4_F32, accumulating over the full D=768 (192 K-chunks,
// split across 4 independent accumulators for ILP). B (sentence chunk) is
// replicated into all 16 columns so every D column is a valid copy of the
// 16 dot products; mask+sum happens per lane, then halves are combined.
__global__ __launch_bounds__(256) void noun_main_kernel(
    const float* __restrict__ tok,    // [F,B,S,D]
    const float* __restrict__ sent,   // [F,B,D]
    const int*   __restrict__ mask,   // [F,B,S]
    float*       __restrict__ partials) // [PAIRS * TILES]
{
    __shared__ __align__(16) float lds_sent[DIM];

    const int wg   = blockIdx.x;
    const int pair = wg >> 2;          // (f*B + b)
    const int quad = wg & 3;
    const int tid  = threadIdx.x;
    const int lane = tid & 31;
    const int wave = tid >> 5;
    const int tile = quad * WAVES_PER_WG + wave;   // 0..31

    // Stage this pair's sentence embedding (3 KB) into LDS once per WG.
    {
        const float* sp = sent + (size_t)pair * DIM;
        for (int i = tid; i < DIM; i += 256) lds_sent[i] = sp[i];
    }
    __syncthreads();

    const int row = lane & 15;         // matrix row M (token within tile)
    const int hi  = lane >> 4;         // 0: lanes 0-15 hold K+0..1, 1: lanes 16-31 hold K+2..3

    // Per-lane A base: exact WMMA A-layout via one global_load_b64 per chunk.
    const float* arow = tok + (size_t)pair * SEQ * DIM
                            + (size_t)(tile * 16 + row) * DIM
                            + (size_t)(hi * 2);
    const float* bptr = lds_sent + hi * 2;

    v8f c0 = {}, c1 = {}, c2 = {}, c3 = {};

    for (int k0 = 0; k0 < DIM; k0 += 16) {
        v2f a0 = *(const v2f*)(arow + k0);
        v2f a1 = *(const v2f*)(arow + k0 + 4);
        v2f a2 = *(const v2f*)(arow + k0 + 8);
        v2f a3 = *(const v2f*)(arow + k0 + 12);
        v2f b0 = *(const v2f*)(bptr + k0);
        v2f b1 = *(const v2f*)(bptr + k0 + 4);
        v2f b2 = *(const v2f*)(bptr + k0 + 8);
        v2f b3 = *(const v2f*)(bptr + k0 + 12);
        // 8 args: (neg_a, A, neg_b, B, c_mod, C, reuse_a, reuse_b)
        c0 = __builtin_amdgcn_wmma_f32_16x16x4_f32(false, a0, false, b0, (short)0, c0, false, false);
        c1 = __builtin_amdgcn_wmma_f32_16x16x4_f32(false, a1, false, b1, (short)0, c1, false, false);
        c2 = __builtin_amdgcn_wmma_f32_16x16x4_f32(false, a2, false, b2, (short)0, c2, false, false);
        c3 = __builtin_amdgcn_wmma_f32_16x16x4_f32(false, a3, false, b3, (short)0, c3, false, false);
    }

    // Each lane now holds dot[tile*16 + hi*8 + r] in (c0+c1+c2+c3)[r], r=0..7
    // (identical across the 16 lanes of its half-wave, since B columns are equal).
    const int mbase = pair * SEQ + tile * 16 + hi * 8;
    float partial = 0.0f;
#pragma unroll
    for (int r = 0; r < 8; ++r) {
        float tr = c0[r] + c1[r] + c2[r] + c3[r];
        partial += mask[mbase + r] ? tr : 0.0f;
    }
    // Combine low-half (M=0..7) and high-half (M=8..15) contributions.
    partial += __shfl_xor(partial, 16, 32);

    if (lane == 0) partials[pair * TILES + tile] = partial;
}

// 128 threads: per-pair T = sum of 32 tile partials (fixed order), apply
// T/(T+eps), deterministic tree reduction, divide by F.
__global__ __launch_bounds__(128) void noun_final_kernel(
    const float* __restrict__ partials, float* __restrict__ out)
{
    __shared__ float red[128];
    const int tid = threadIdx.x;   // pair index 0..127
    float t = 0.0f;
#pragma unroll
    for (int j = 0; j < TILES; ++j) t += partials[tid * TILES + j];
    red[tid] = t / (t + 1e-9f);
    __syncthreads();
#pragma unroll
    for (int s = 64; s > 0; s >>= 1) {
        if (tid < s) red[tid] += red[tid + s];
        __syncthreads();
    }
    if (tid == 0) out[0] = red[0] * (1.0f / (float)FDIM);
}

extern "C" void kernel_launch(void* const* d_in, const int* in_sizes, int n_in,
                              void* d_out, int out_size, void* d_ws, size_t ws_size,
                              hipStream_t stream) {
    const float* tok  = (const float*)d_in[0];   // [2,64,512,768] f32
    const float* sent = (const float*)d_in[1];   // [2,64,768]     f32
    const int*   mask = (const int*)d_in[2];     // [2,64,512]     i32
    float* out      = (float*)d_out;             // scalar f32
    float* partials = (float*)d_ws;              // PAIRS*TILES floats = 16 KB

    noun_main_kernel<<<PAIRS * WGS_PER_PAIR, 256, 0, stream>>>(tok, sent, mask, partials);
    noun_final_kernel<<<1, 128, 0, stream>>>(partials, out);
}